// RGCNLayerScratch_72799695667854
// MI455X (gfx1250) — compile-verified
//
#include <hip/hip_runtime.h>

#define N_NODES 50000
#define N_EDGES 800000
#define N_REL   8
#define N_BASES 16
#define D       128

typedef __attribute__((ext_vector_type(2))) float v2f;
typedef __attribute__((ext_vector_type(8))) float v8f;

// ---------------------------------------------------------------------------
// W[r] = sum_b coeff[r,b] * bases[b]   -> [8,128,128]
// ---------------------------------------------------------------------------
__global__ void make_W_kernel(const float* __restrict__ bases,
                              const float* __restrict__ coeff,
                              float* __restrict__ W) {
    int idx = blockIdx.x * 256 + threadIdx.x;        // 8 * 16384
    if (idx >= N_REL * D * D) return;
    int r  = idx >> 14;          // / 16384
    int io = idx & 16383;
    float acc = 0.0f;
#pragma unroll
    for (int b = 0; b < N_BASES; ++b)
        acc += coeff[r * N_BASES + b] * bases[b * D * D + io];
    W[idx] = acc;
}

// ---------------------------------------------------------------------------
// degree helpers
// ---------------------------------------------------------------------------
__global__ void zero_f32_kernel(float* __restrict__ p, int n) {
    int i = blockIdx.x * 256 + threadIdx.x;
    if (i < n) p[i] = 0.0f;
}

__global__ void count_deg_kernel(const int* __restrict__ dst,
                                 const int* __restrict__ etype,
                                 float* __restrict__ deg) {
    int e = blockIdx.x * 256 + threadIdx.x;
    if (e < N_EDGES) atomicAdd(&deg[etype[e] * N_NODES + dst[e]], 1.0f);
}

__global__ void inv_deg_kernel(float* __restrict__ deg, int n) {
    int i = blockIdx.x * 256 + threadIdx.x;
    if (i < n) deg[i] = 1.0f / fmaxf(deg[i], 1.0f);
}

// ---------------------------------------------------------------------------
// Y[m0:m0+16, :] = X[m0:m0+16, :] @ Wm   (128x128 Wm), fp32 WMMA 16x16x4
// block = 256 threads = 8 waves; wave w owns N-tile [16w, 16w+16)
// ---------------------------------------------------------------------------
__global__ __launch_bounds__(256) void rgcn_gemm_kernel(
        const float* __restrict__ X,
        const float* __restrict__ Wm,
        float* __restrict__ Y) {
    __shared__ float As[16 * D];                     // 8 KB A tile

    const int tid  = threadIdx.x;
    const int wave = tid >> 5;
    const int lane = tid & 31;
    const int m0   = blockIdx.x * 16;

    // stage A tile: 2048 floats, 256 threads x 2 float4
    {
        const float4* src4 = (const float4*)(X + (size_t)m0 * D);
        float4* dst4 = (float4*)As;
        dst4[tid]       = src4[tid];
        dst4[tid + 256] = src4[tid + 256];
    }
    __syncthreads();

    const int row   = lane & 15;        // A row / D column
    const int col   = lane & 15;        // B/C column
    const int khalf = lane >> 4;        // 0: K={k,k+1}  1: K={k+2,k+3}
    const int n0    = wave * 16;

    v8f c = {};
#pragma unroll 4
    for (int k = 0; k < D; k += 4) {
        const int kk = k + khalf * 2;
        v2f a, b;
        a[0] = As[row * D + kk];
        a[1] = As[row * D + kk + 1];
        b[0] = Wm[(size_t)kk * D + n0 + col];
        b[1] = Wm[(size_t)(kk + 1) * D + n0 + col];
        c = __builtin_amdgcn_wmma_f32_16x16x4_f32(
                /*neg_a=*/false, a, /*neg_b=*/false, b,
                /*c_mod=*/(short)0, c, /*reuse_a=*/false, /*reuse_b=*/false);
    }

    // store D matrix: VGPR v -> row (v + 8*khalf), col = lane&15
    float* outp = Y + (size_t)m0 * D + n0;
#pragma unroll
    for (int v = 0; v < 8; ++v) {
        int r = v + khalf * 8;
        outp[(size_t)r * D + col] = c[v];
    }
}

// ---------------------------------------------------------------------------
// for edges of relation `rel`: out[dst] += slab[src] * deg_inv[rel*N + dst]
// one warp per edge; lane handles cols {l, l+32, l+64, l+96}
// ---------------------------------------------------------------------------
__global__ __launch_bounds__(256) void rgcn_scatter_kernel(
        const float* __restrict__ slab,
        const int* __restrict__ src,
        const int* __restrict__ dst,
        const int* __restrict__ etype,
        const float* __restrict__ deg_inv,
        float* __restrict__ out,
        int rel) {
    int e = blockIdx.x * 8 + (threadIdx.x >> 5);
    if (e >= N_EDGES) return;
    if (etype[e] != rel) return;
    const int lane = threadIdx.x & 31;
    const int s = src[e];
    const int d = dst[e];
    const float w = deg_inv[rel * N_NODES + d];
    const float* sp = slab + (size_t)s * D;
    float* op = out + (size_t)d * D;
#pragma unroll
    for (int j = 0; j < 4; ++j) {
        int cidx = lane + j * 32;
        atomicAdd(op + cidx, sp[cidx] * w);
    }
}

// ---------------------------------------------------------------------------
extern "C" void kernel_launch(void* const* d_in, const int* in_sizes, int n_in,
                              void* d_out, int out_size, void* d_ws, size_t ws_size,
                              hipStream_t stream) {
    const float* x         = (const float*)d_in[0];  // [N,128]
    const float* bases     = (const float*)d_in[1];  // [16,128,128]
    const float* coeff     = (const float*)d_in[2];  // [8,16]
    const float* self_loop = (const float*)d_in[3];  // [128,128]
    const int*   edge_idx  = (const int*)d_in[4];    // [2,E]
    const int*   edge_type = (const int*)d_in[5];    // [E]
    const int*   srcp = edge_idx;
    const int*   dstp = edge_idx + N_EDGES;
    float* out = (float*)d_out;

    // workspace layout (floats): W[8*128*128] | deg[8*N] | slab[N*128]  (~28 MB)
    float* ws   = (float*)d_ws;
    float* Wall = ws;
    float* deg  = Wall + N_REL * D * D;
    float* slab = deg + N_REL * N_NODES;

    // 1) combine bases into per-relation weights
    make_W_kernel<<<(N_REL * D * D + 255) / 256, 256, 0, stream>>>(bases, coeff, Wall);

    // 2) per-(relation,dst) in-degree -> reciprocal
    zero_f32_kernel<<<(N_REL * N_NODES + 255) / 256, 256, 0, stream>>>(deg, N_REL * N_NODES);
    count_deg_kernel<<<(N_EDGES + 255) / 256, 256, 0, stream>>>(dstp, edge_type, deg);
    inv_deg_kernel<<<(N_REL * N_NODES + 255) / 256, 256, 0, stream>>>(deg, N_REL * N_NODES);

    // 3) out = x @ self_loop (initializes full output)
    const int mtiles = N_NODES / 16;   // 3125, exact
    rgcn_gemm_kernel<<<mtiles, 256, 0, stream>>>(x, self_loop, out);

    // 4) per relation: slab = x @ W[r]; out[dst] += slab[src] * deg_inv
    for (int r = 0; r < N_REL; ++r) {
        rgcn_gemm_kernel<<<mtiles, 256, 0, stream>>>(x, Wall + (size_t)r * D * D, slab);
        rgcn_scatter_kernel<<<(N_EDGES + 7) / 8, 256, 0, stream>>>(
            slab, srcp, dstp, edge_type, deg, out, r);
    }
}